// BuildCostVolume_43258910605836
// MI455X (gfx1250) — compile-verified
//
#include <hip/hip_runtime.h>

// ---------------------------------------------------------------------------
// Problem constants (from reference): A=9, HW=48, input spatial 432=9*48,
// B=4, CIN=COUT=64, disparities -4..4, 17 half-step disps x 4 repeats (D=68).
// ---------------------------------------------------------------------------
#define ANG     9
#define HW      48
#define CIN     64
#define COUT    64
#define IN_HW   432
#define PLANE   (IN_HW * IN_HW)     // 186624 elements per (b,ci) plane
#define WTAPS   81                  // 9*9 weight taps
#define NDISP   68
#define BATCH   4

typedef __attribute__((ext_vector_type(2))) float v2f;
typedef __attribute__((ext_vector_type(4))) float v4f;
typedef __attribute__((ext_vector_type(8))) float v8f;

// ---------------------------------------------------------------------------
// Implicit-GEMM conv accumulation for one disparity d, one output row `oh`,
// one batch plane `xb`, one 16-wide cout stripe (co0), across all 48 ow.
// Uses V_WMMA_F32_16X16X4_F32 rank-4 updates over K = (tap, ci-quad).
//
// Tap geometry (derived, kernel flipped for d>0):
//   input row  = 9*rr + kh,  rr = oh - d*(kh-4)   (skip whole kh if rr OOB)
//   input col  = 9*cc + kw,  cc = ow - d*(kw-4)   (per-lane validity)
//   weight tap = (flip? 8-kh : kh, flip? 8-kw : kw)
// ---------------------------------------------------------------------------
__device__ __forceinline__ void conv_accum(const float* __restrict__ xb,
                                           const float* __restrict__ wgt,
                                           int oh, int d, int co0, int lane,
                                           v8f acc[3])
{
    const int m     = lane & 15;   // N index (ow within 16-tile) / M index for A
    const int khalf = lane >> 4;   // selects K pair {0,1} vs {2,3}
    const bool flip = (d > 0);

    // Per-wave weight row pointer: element (co0+m, ci=kA, tap) = wp[kA*81 + wtap]
    const float* __restrict__ wrow = wgt + (size_t)(co0 + m) * (CIN * WTAPS);

    #pragma unroll 1
    for (int kh = 0; kh < 9; ++kh) {
        const int rr = oh - d * (kh - 4);
        if ((unsigned)rr >= (unsigned)HW) continue;          // uniform: EXEC stays full
        const int rowbase = (9 * rr + kh) * IN_HW;
        const int wh = flip ? (8 - kh) : kh;

        #pragma unroll 1
        for (int kw = 0; kw < 9; ++kw) {
            const int sw   = d * (kw - 4);
            const int ww   = flip ? (8 - kw) : kw;
            const int wtap = wh * 9 + ww;

            // Per-lane column addresses for the 3 N-tiles (clamped when invalid)
            int  col[3];
            bool ok[3];
            #pragma unroll
            for (int t = 0; t < 3; ++t) {
                const int ow = 16 * t + m;
                const int cc = ow - sw;
                ok[t]  = ((unsigned)cc < (unsigned)HW);
                col[t] = ok[t] ? (rowbase + 9 * cc + kw) : 0;
            }

            const float* __restrict__ wp = wrow + wtap;

            #pragma unroll 4
            for (int ci0 = 0; ci0 < CIN; ci0 += 4) {
                const int kA = ci0 + 2 * khalf;              // K index pair base
                // A fragment 16x4 f32: VGPR0 = K{0,2}, VGPR1 = K{1,3} by lane half
                v2f a;
                a.x = wp[(size_t)kA * WTAPS];
                a.y = wp[(size_t)(kA + 1) * WTAPS];

                const float* __restrict__ bp = xb + (size_t)kA * PLANE;
                #pragma unroll
                for (int t = 0; t < 3; ++t) {
                    const float b0 = bp[col[t]];
                    const float b1 = bp[(size_t)PLANE + col[t]];
                    v2f bf;
                    bf.x = ok[t] ? b0 : 0.0f;
                    bf.y = ok[t] ? b1 : 0.0f;
                    acc[t] = __builtin_amdgcn_wmma_f32_16x16x4_f32(
                        /*neg_a=*/false, a, /*neg_b=*/false, bf,
                        /*c_mod=*/(short)0, acc[t],
                        /*reuse_a=*/false, /*reuse_b=*/false);
                }
            }
        }
    }
}

// ---------------------------------------------------------------------------
// Main kernel: block = (oh, b, j). j in [0,8) selects the half-integer
// disparity pair (d_lo=j-4, d_hi=j-3). Each of 4 waves owns a 16-cout stripe.
// Computes both convs in-register, blends 0.5*(lo+hi), stores 4 D-replicas.
// ---------------------------------------------------------------------------
__global__ __launch_bounds__(128) void cost_volume_wmma(
    const float* __restrict__ x,     // [4,64,432,432]
    const float* __restrict__ wgt,   // [64,64,9,9]
    float* __restrict__ out)         // [4,64,68,48,48]
{
    const int oh   = blockIdx.x;         // 0..47
    const int b    = blockIdx.y;         // 0..3
    const int j    = blockIdx.z;         // 0..7
    const int wv   = threadIdx.x >> 5;
    const int lane = threadIdx.x & 31;
    const int co0  = wv * 16;

    const float* __restrict__ xb = x + (size_t)b * CIN * PLANE;

    v8f accL[3], accH[3];
    #pragma unroll
    for (int t = 0; t < 3; ++t)
        #pragma unroll
        for (int i = 0; i < 8; ++i) { accL[t][i] = 0.0f; accH[t][i] = 0.0f; }

    conv_accum(xb, wgt, oh, j - 4, co0, lane, accL);
    conv_accum(xb, wgt, oh, j - 3, co0, lane, accH);

    // Blend and store. C/D layout: VGPR g -> M = g (lanes 0-15) / g+8 (16-31),
    // N = lane&15. Disparity index i = 2j+1 occupies D = 8j+4 .. 8j+7.
    const int m     = lane & 15;
    const int khalf = lane >> 4;
    const int Dbase = 8 * j + 4;

    #pragma unroll
    for (int t = 0; t < 3; ++t) {
        const int ow = 16 * t + m;
        #pragma unroll
        for (int g = 0; g < 8; ++g) {
            const float v  = 0.5f * (accL[t][g] + accH[t][g]);
            const int   co = co0 + g + 8 * khalf;
            float* p = out + (((((size_t)b * COUT + co) * NDISP + Dbase) * HW + oh) * HW + ow);
            __builtin_nontemporal_store(v, p);
            __builtin_nontemporal_store(v, p + (size_t)HW * HW);
            __builtin_nontemporal_store(v, p + (size_t)2 * HW * HW);
            __builtin_nontemporal_store(v, p + (size_t)3 * HW * HW);
        }
    }
}

// ---------------------------------------------------------------------------
// Streaming zero-fill: the 36 integer-disparity D-slots are exactly zero
// (blend weights are 0 in the reference); simplest is to zero everything
// first, then overwrite the 32 blended slots above.
// ---------------------------------------------------------------------------
__global__ __launch_bounds__(256) void zero_fill(v4f* __restrict__ out, int n4)
{
    const int i = blockIdx.x * blockDim.x + threadIdx.x;
    if (i < n4) {
        v4f z;
        z.x = 0.0f; z.y = 0.0f; z.z = 0.0f; z.w = 0.0f;
        __builtin_nontemporal_store(z, out + i);
    }
}

extern "C" void kernel_launch(void* const* d_in, const int* in_sizes, int n_in,
                              void* d_out, int out_size, void* d_ws, size_t ws_size,
                              hipStream_t stream)
{
    const float* x   = (const float*)d_in[0];   // [4,64,432,432] f32
    const float* wgt = (const float*)d_in[1];   // [64,64,9,9] f32
    float* out = (float*)d_out;                 // [4,64,68,48,48] f32

    // 1) zero entire output (covers all integer-disparity slots exactly)
    const int n4 = out_size / 4;                // out_size = 40,108,032, divisible by 4
    zero_fill<<<(n4 + 255) / 256, 256, 0, stream>>>((v4f*)out, n4);

    // 2) WMMA implicit-GEMM convs + blend, written directly to output
    dim3 grid(HW, BATCH, 8);                    // (oh, b, disparity-pair)
    cost_volume_wmma<<<grid, 128, 0, stream>>>(x, wgt, out);
}